// CTCLayer_5291399709503
// MI455X (gfx1250) — compile-verified
//
#include <hip/hip_runtime.h>
#include <hip/hip_bf16.h>

#ifndef __has_builtin
#define __has_builtin(x) 0
#endif

// gfx1250 async global->LDS path (guarded: falls back to register pipelining
// + global_prefetch_b8 if the builtin is not declared by this toolchain).
#if __has_builtin(__builtin_amdgcn_global_load_async_to_lds_b128) && \
    __has_builtin(__builtin_amdgcn_s_wait_asynccnt)
#define CTC_USE_ASYNC 1
#else
#define CTC_USE_ASYNC 0
#endif

#define CTC_AS_G __attribute__((address_space(1)))
#define CTC_AS_L __attribute__((address_space(3)))

// Matches the builtin's parameter type: int __vector(4) (16 bytes).
typedef int v4i_t __attribute__((__vector_size__(4 * sizeof(int))));

namespace {

constexpr int kB = 256, kT = 512, kC = 128, kL = 64;
constexpr int kBlank = kC - 1;          // 127
constexpr float kNeg = -1e30f;          // -inf surrogate (log2 domain too)
constexpr float kEps = 1e-7f;
constexpr float kLn2 = 0.69314718055994530942f;

// Native base-2 transcendentals: v_exp_f32 / v_log_f32 (log2 on AMD HW).
__device__ __forceinline__ float hw_exp2(float x) {
  return __builtin_amdgcn_exp2f(x);
}
__device__ __forceinline__ float hw_log2(float x) {
  return __builtin_amdgcn_logf(x);
}

// 3-way logaddexp in LOG2 domain (base-2 HW ops, so no ln2-conversion
// multiplies sit on the dependency chain).
// z is pre-masked to kNeg when the skip transition is absent.
// All-kNeg case: m=-1e30, sum=3, m+log2(3) == -1e30 in f32 -> matches ref.
__device__ __forceinline__ float lae3_2(float x, float y, float z) {
  float m = fmaxf(fmaxf(x, y), z);
  return m + hw_log2(hw_exp2(x - m) + hw_exp2(y - m) + hw_exp2(z - m));
}

// alpha[s-1] for s = lane + 32k: lane-1 of same register, or lane 31 of the
// previous register (k-1) for lane 0. `prev` is kNeg for k==0.
__device__ __forceinline__ float shift1(float cur, float prev, int lane) {
  float u = __shfl_up(cur, 1, 32);
  float b = __shfl(prev, 31, 32);
  return (lane == 0) ? b : u;
}
// alpha[s-2]: lane-2 of same register, or lanes 30/31 of previous register.
__device__ __forceinline__ float shift2(float cur, float prev, int lane) {
  float u = __shfl_up(cur, 2, 32);
  float b = __shfl(prev, 30 + (lane & 1), 32);
  return (lane < 2) ? b : u;
}

// Read alpha[s]; k = s>>5 is wave-uniform (label_length is uniform per block).
__device__ __forceinline__ float read_state(int s, float a0, float a1, float a2,
                                            float a3, float a4) {
  int k = s >> 5, l = s & 31;
  float x = (k == 0) ? a0 : (k == 1) ? a1 : (k == 2) ? a2 : (k == 3) ? a3 : a4;
  return __shfl(x, l, 32);
}

}  // namespace

__global__ __launch_bounds__(32) void ctc_loss_kernel(
    const int* __restrict__ y_true, const float* __restrict__ y_pred,
    const int* __restrict__ in_len_p, const int* __restrict__ lab_len_p,
    float* __restrict__ out) {
  const int b = (int)blockIdx.x;
  const int lane = (int)threadIdx.x;  // wave32: one wave per batch element
  const int* yt = y_true + b * kL;
  const float* P = y_pred + (size_t)b * (kT * kC);
  const int in_len = in_len_p[b];   // wave-uniform
  const int lab = lab_len_p[b];     // wave-uniform

  // Extended-label column + skip flag for states s = lane + 32k, k=0..3.
  // Even s -> blank(127), skip=false. Odd s -> label (s-1)/2,
  // skip = (s>=3) && label != previous label.
  int col[4];
  bool skp[4];
#pragma unroll
  for (int k = 0; k < 4; ++k) {
    int s = lane + 32 * k;
    if (s & 1) {
      int i = (s - 1) >> 1;
      int yi = yt[i];
      col[k] = yi;
      skp[k] = (s >= 3) && (yi != yt[i - 1]);
    } else {
      col[k] = kBlank;
      skp[k] = false;
    }
  }

  // alpha registers (log2 domain): a<k> = alpha[lane + 32k]; a4 = s=128.
  float a0 = kNeg, a1 = kNeg, a2 = kNeg, a3 = kNeg, a4 = kNeg;
  {
    // t = 0: only s=0 (blank) and s=1 (first label) are reachable.
    float v0 = P[col[0]];
    float i0 = hw_log2(v0 + kEps);
    a0 = (lane < 2) ? i0 : kNeg;
  }

  // One recurrence step given the 5 gathered probabilities for this t.
  auto step = [&](float v0, float v1, float v2, float v3, float vb) {
    float lp0 = hw_log2(v0 + kEps), lp1 = hw_log2(v1 + kEps);
    float lp2 = hw_log2(v2 + kEps), lp3 = hw_log2(v3 + kEps);
    float lpb = hw_log2(vb + kEps);
    float s10 = shift1(a0, kNeg, lane), s20 = shift2(a0, kNeg, lane);
    float s11 = shift1(a1, a0, lane), s21 = shift2(a1, a0, lane);
    float s12 = shift1(a2, a1, lane), s22 = shift2(a2, a1, lane);
    float s13 = shift1(a3, a2, lane), s23 = shift2(a3, a2, lane);
    float s14 = __shfl(a3, 31, 32);  // alpha[127] broadcast for s=128
    float n0 = lae3_2(a0, s10, skp[0] ? s20 : kNeg) + lp0;
    float n1 = lae3_2(a1, s11, skp[1] ? s21 : kNeg) + lp1;
    float n2 = lae3_2(a2, s12, skp[2] ? s22 : kNeg) + lp2;
    float n3 = lae3_2(a3, s13, skp[3] ? s23 : kNeg) + lp3;
    float n4 = lae3_2(a4, s14, kNeg) + lpb;  // s=128 is blank: no skip
    a0 = n0; a1 = n1; a2 = n2; a3 = n3; a4 = n4;
  };

  const int tmax = in_len - 1;  // reference freezes alpha for t >= in_len

#if CTC_USE_ASYNC
  // 8-deep LDS ring of y_pred rows (512 B each; 4 KB total), filled by async
  // copies issued 7 steps ahead; LDS->register gathers pre-fetched 1 step
  // ahead so neither HBM/L2 nor LDS latency sits on the recurrence chain.
  // Row r lives in buf (r-1)&7.
  __shared__ __attribute__((aligned(16))) float rowbuf[8][kC];
  auto issue = [&](int buf, int trow) {
    CTC_AS_G v4i_t* g = (CTC_AS_G v4i_t*)(P + (size_t)trow * kC) + lane;
    CTC_AS_L v4i_t* l = (CTC_AS_L v4i_t*)(&rowbuf[buf][0]) + lane;
    __builtin_amdgcn_global_load_async_to_lds_b128(g, l, 0, 0);
  };
  auto clampt = [&](int t) { return (t > tmax) ? tmax : t; };
  if (tmax >= 1) {
#pragma unroll
    for (int r = 1; r <= 7; ++r) issue(r - 1, clampt(r));
    __builtin_amdgcn_s_wait_asynccnt(6);  // row 1 resident
    asm volatile("" ::: "memory");
    float w0, w1, w2, w3, wb;
    {
      const float* r = rowbuf[0];
      w0 = r[col[0]]; w1 = r[col[1]]; w2 = r[col[2]]; w3 = r[col[3]];
      wb = r[kBlank];
    }
    for (int t = 1; t <= tmax; ++t) {
      float v0 = w0, v1 = w1, v2 = w2, v3 = w3, vb = wb;
      asm volatile("" ::: "memory");
      issue((t + 6) & 7, clampt(t + 7));       // row t+7
      __builtin_amdgcn_s_wait_asynccnt(6);     // rows <= t+1 resident
      asm volatile("" ::: "memory");
      {
        const float* r = rowbuf[t & 7];        // row t+1 (clamped at tail)
        w0 = r[col[0]]; w1 = r[col[1]]; w2 = r[col[2]]; w3 = r[col[3]];
        wb = r[kBlank];
      }
      step(v0, v1, v2, v3, vb);
    }
  }
#else
  // Fallback: register double-buffer of the 5 gathers + prefetch 12 rows ahead
  // (emits gfx1250 global_prefetch_b8).
  if (tmax >= 1) {
    const float* r1 = P + kC;
    float w0 = r1[col[0]], w1 = r1[col[1]], w2 = r1[col[2]], w3 = r1[col[3]];
    float wb = r1[kBlank];
    for (int t = 1; t <= tmax; ++t) {
      float v0 = w0, v1 = w1, v2 = w2, v3 = w3, vb = wb;
      int tn = (t < tmax) ? t + 1 : t;
      const float* rn = P + (size_t)tn * kC;
      if (lane < 4) {
        int tp = (t + 12 <= tmax) ? t + 12 : tmax;
        __builtin_prefetch(P + (size_t)tp * kC + lane * 32, 0, 3);
      }
      w0 = rn[col[0]]; w1 = rn[col[1]]; w2 = rn[col[2]]; w3 = rn[col[3]];
      wb = rn[kBlank];
      step(v0, v1, v2, v3, vb);
    }
  }
#endif

  // loss = -logaddexp(alpha[2*lab], alpha[2*lab - 1]) ; convert log2 -> ln
  int sl = 2 * lab;  // lab in [32,64] -> sl in [64,128]
  float vl = read_state(sl, a0, a1, a2, a3, a4);
  float vp = read_state(sl - 1, a0, a1, a2, a3, a4);
  float loss = -kLn2 * lae3_2(vl, vp, kNeg);
  if (lane == 0) out[b] = loss;
}

extern "C" void kernel_launch(void* const* d_in, const int* in_sizes, int n_in,
                              void* d_out, int out_size, void* d_ws,
                              size_t ws_size, hipStream_t stream) {
  (void)in_sizes; (void)n_in; (void)out_size; (void)d_ws; (void)ws_size;
  const int* y_true = (const int*)d_in[0];        // [B, L] int32
  const float* y_pred = (const float*)d_in[1];    // [B, T, C] float32
  const int* input_length = (const int*)d_in[2];  // [B, 1] int32
  const int* label_length = (const int*)d_in[3];  // [B, 1] int32
  float* out = (float*)d_out;                     // [B, 1] float32

  ctc_loss_kernel<<<kB, 32, 0, stream>>>(y_true, y_pred, input_length,
                                         label_length, out);
}